// GCN_diff_4861902979196
// MI455X (gfx1250) — compile-verified
//
#include <hip/hip_runtime.h>
#include <hip/hip_bf16.h>

typedef float v2f __attribute__((ext_vector_type(2)));
typedef float v8f __attribute__((ext_vector_type(8)));

#define D 128

// Non-returning HW float atomic (GLOBAL_ATOMIC_ADD_F32, STOREcnt-tracked).
// Guarantees no CAS loop regardless of fp-atomic lowering policy.
__device__ __forceinline__ void atomic_add_f32(float* p, float v) {
    asm volatile("global_atomic_add_f32 %0, %1, off"
                 :
                 : "v"(p), "v"(v)
                 : "memory");
}

// ---------------- degree / normalization ----------------

__global__ void k_deg_init(float* __restrict__ dinv, int N) {
    int i = blockIdx.x * blockDim.x + threadIdx.x;
    if (i < N) dinv[i] = 1.0f;   // self-loop contribution to degree
}

__global__ void k_deg_count(const long long* __restrict__ ei, float* __restrict__ dinv,
                            long long E) {
    long long e = (long long)blockIdx.x * blockDim.x + threadIdx.x;
    if (e < E) {
        long long c = ei[E + e];       // target node
        atomic_add_f32(&dinv[c], 1.0f);
    }
}

__global__ void k_deg_rsqrt(float* __restrict__ dinv, int N) {
    int i = blockIdx.x * blockDim.x + threadIdx.x;
    if (i < N) dinv[i] = rsqrtf(dinv[i]);   // deg >= 1 always (self loops)
}

// ---------------- GEMM: h = x @ W^T via V_WMMA_F32_16X16X4_F32 ----------------
// One wave computes a 16x16 tile of h. 8 waves per block -> 16 rows x 128 cols.
// A (16x4 f32): lane m=lane&15 is row M; kh=lane>>4 selects K pair {2kh, 2kh+1}.
// B (4x16 f32): lane m is col N; same K-pair split. B[k][n] = W[n][k].
// C/D (16x16 f32, 8 VGPRs): N = lane&15, M = r + 8*kh.

__global__ __launch_bounds__(256) void k_gemm(const float* __restrict__ x,
                                              const float* __restrict__ w,
                                              float* __restrict__ h) {
    __shared__ float xs[16 * D];   // 8 KB x-tile shared by all 8 waves

    const int tid  = threadIdx.x;
    const int wave = tid >> 5;       // 0..7 -> 16-column slice
    const int lane = tid & 31;
    const int row0 = blockIdx.x * 16;

    // cooperative load of 16x128 x-tile (2048 floats = 512 float4, 2 per thread)
    {
        const float4* src = (const float4*)(x + (size_t)row0 * D);
        float4* dst = (float4*)xs;
        dst[tid]       = src[tid];
        dst[tid + 256] = src[tid + 256];
    }
    __syncthreads();

    const int m   = lane & 15;
    const int kh  = lane >> 4;           // which K pair of the 4-wide slab
    const int col = wave * 16 + m;       // output feature column 0..127

    v8f c = {};
    #pragma unroll
    for (int k = 0; k < D; k += 4) {
        v2f a = *(const v2f*)(&xs[m * D + k + 2 * kh]);
        // B[k][col] = W[col][k]  (w is row-major [D][D])
        v2f b = *(const v2f*)(&w[(size_t)col * D + k + 2 * kh]);
        c = __builtin_amdgcn_wmma_f32_16x16x4_f32(
                /*neg_a=*/false, a, /*neg_b=*/false, b,
                /*c_mod=*/(short)0, c, /*reuse_a=*/false, /*reuse_b=*/false);
    }

    // store D-matrix: h[row0 + r + 8*kh][col]
    float* hp = h + (size_t)row0 * D + col;
    #pragma unroll
    for (int r = 0; r < 8; ++r) {
        hp[(size_t)(r + 8 * kh) * D] = c[r];
    }
}

// ---------------- self-loop init: out[i] = h[i] * dinv[i]^2 ----------------
// One thread per (node, 4-feature chunk); fully overwrites poisoned d_out.

__global__ void k_self(const float* __restrict__ h, const float* __restrict__ dinv,
                       float* __restrict__ out, int N) {
    int t = blockIdx.x * blockDim.x + threadIdx.x;   // N*32 units
    if (t >= N * 32) return;
    int i = t >> 5;
    int j = t & 31;
    float s = dinv[i];
    s = s * s;
    float4 v = ((const float4*)(h + (size_t)i * D))[j];
    v.x *= s; v.y *= s; v.z *= s; v.w *= s;
    ((float4*)(out + (size_t)i * D))[j] = v;
}

// ---------------- edge scatter: out[col] += h[row] * dinv[row]*dinv[col] ----------------
// One wave per edge (32 lanes x 4 features = 128): coalesced float4 gather of
// L2-resident h row + 4 non-returning HW f32 atomics per lane into out row.

__global__ __launch_bounds__(256) void k_scatter(const long long* __restrict__ ei,
                                                 const float* __restrict__ h,
                                                 const float* __restrict__ dinv,
                                                 float* __restrict__ out,
                                                 long long E) {
    long long t = (long long)blockIdx.x * blockDim.x + threadIdx.x;  // E*32 units
    long long e = t >> 5;
    if (e >= E) return;
    int j = (int)(t & 31);
    long long r = ei[e];         // source
    long long c = ei[E + e];     // target
    float coeff = dinv[r] * dinv[c];
    float4 hv = ((const float4*)(h + (size_t)r * D))[j];
    float* op = out + (size_t)c * D + (size_t)j * 4;
    atomic_add_f32(op + 0, hv.x * coeff);
    atomic_add_f32(op + 1, hv.y * coeff);
    atomic_add_f32(op + 2, hv.z * coeff);
    atomic_add_f32(op + 3, hv.w * coeff);
}

// ---------------- epilogue: out = relu(out + bias) ----------------

__global__ void k_bias_relu(float* __restrict__ out, const float* __restrict__ bias, int N) {
    int t = blockIdx.x * blockDim.x + threadIdx.x;   // N*32 units
    if (t >= N * 32) return;
    int j = t & 31;
    const float4 b = ((const float4*)bias)[j];
    float4 v = ((float4*)out)[t];
    v.x = fmaxf(v.x + b.x, 0.0f);
    v.y = fmaxf(v.y + b.y, 0.0f);
    v.z = fmaxf(v.z + b.z, 0.0f);
    v.w = fmaxf(v.w + b.w, 0.0f);
    ((float4*)out)[t] = v;
}

// ---------------- launcher ----------------

extern "C" void kernel_launch(void* const* d_in, const int* in_sizes, int n_in,
                              void* d_out, int out_size, void* d_ws, size_t ws_size,
                              hipStream_t stream) {
    const float*     x    = (const float*)d_in[0];
    const long long* ei   = (const long long*)d_in[1];   // int64 [2, E]
    const float*     w    = (const float*)d_in[2];       // [D, D]
    const float*     bias = (const float*)d_in[3];       // [D]
    float*           out  = (float*)d_out;               // [N, D]

    const int       N = in_sizes[0] / D;                 // 100000
    const long long E = (long long)(in_sizes[1] / 2);    // 1600000

    // workspace layout: h [N*D floats] then dinv [N floats]
    float* h    = (float*)d_ws;
    float* dinv = h + (size_t)N * D;

    const int T = 256;

    // 1-3: symmetric normalization coefficients
    k_deg_init <<<(N + T - 1) / T, T, 0, stream>>>(dinv, N);
    k_deg_count<<<(unsigned)((E + T - 1) / T), T, 0, stream>>>(ei, dinv, E);
    k_deg_rsqrt<<<(N + T - 1) / T, T, 0, stream>>>(dinv, N);

    // 4: h = x @ W^T  (WMMA, one block = 16 rows x 128 cols)
    k_gemm<<<N / 16, T, 0, stream>>>(x, w, h);

    // 5: self-loop term initializes out
    k_self<<<(N * 32 + T - 1) / T, T, 0, stream>>>(h, dinv, out, N);

    // 6: edge aggregation
    long long su = E * 32;
    k_scatter<<<(unsigned)((su + T - 1) / T), T, 0, stream>>>(ei, h, dinv, out, E);

    // 7: bias + relu
    k_bias_relu<<<(N * 32 + T - 1) / T, T, 0, stream>>>(out, bias, N);
}